// QuantBatchedEmbeddingBag_71253507440834
// MI455X (gfx1250) — compile-verified
//
#include <hip/hip_runtime.h>
#include <hip/hip_bf16.h>

// Problem constants (match reference)
#define TT 8
#define RR 200000
#define DD 128
#define BB 2048
#define ROWB (DD * 4)       // 512 bytes per row (codes stored as int32)
#define WAVES 4             // waves (bags) per block
#define CH 16               // rows per async chunk (double buffered)

// One wave (32 lanes) per bag. Lane l owns dims [4l, 4l+3].
// Rows are DMA'd global->LDS with CDNA5 async loads (ASYNCcnt), double buffered.
__global__ __launch_bounds__(WAVES * 32)
void tbe_int8_pool_kernel(const int* __restrict__ indices,
                          const int* __restrict__ offsets,
                          const int* __restrict__ qweights,
                          const float* __restrict__ scale_shift,
                          float* __restrict__ out) {
    // WAVES waves * 2 buffers * CH rows * 512B = 64 KB static LDS
    __shared__ int4 smem[WAVES * 2 * CH * (ROWB / 16)];

    const int lane = threadIdx.x & 31;
    const int wv   = threadIdx.x >> 5;
    const int bag  = blockIdx.x * WAVES + wv;
    if (bag >= TT * BB) return;

    const int start = offsets[bag];
    const int end   = offsets[bag + 1];
    const int n     = end - start;
    const int t     = bag / BB;
    const int b     = bag - t * BB;

    const char*   qbase  = (const char*)qweights + (size_t)t * RR * ROWB;
    const float2* ssbase = (const float2*)scale_shift + (size_t)t * RR;

    // Raw LDS byte offset of this wave's staging area: low 32 bits of the
    // generic pointer to shared memory == group-segment byte offset.
    const unsigned lds_wave =
        (unsigned)(uintptr_t)(smem) + (unsigned)(wv * 2 * CH * ROWB);

    float4 acc = {0.f, 0.f, 0.f, 0.f};
    float  shift_sum = 0.f;

    const int nch = (n + CH - 1) / CH;

    int    idx_cur = 0;
    float2 ss_cur  = {0.f, 0.f};

    if (nch > 0) {
        // Prefetch chunk-0 metadata: lane j holds index/scale-shift of row j.
        const int m0 = min(CH, n);
        if (lane < m0) {
            idx_cur = indices[start + lane];
            ss_cur  = ssbase[idx_cur];
        }
        // Issue chunk 0 into buffer 0: one async 512B row per iteration
        // (32 lanes x 16B each), no VGPR landing registers consumed.
        for (int j = 0; j < m0; ++j) {
            const int idxj = __shfl(idx_cur, j, 32);
            const char* g  = qbase + (size_t)idxj * ROWB + lane * 16;
            const unsigned l = lds_wave + (unsigned)(j * ROWB + lane * 16);
            asm volatile("global_load_async_to_lds_b128 %0, %1, off"
                         :: "v"(l), "v"(g) : "memory");
        }
    }

    for (int c = 0; c < nch; ++c) {
        const int  m         = min(CH, n - c * CH);
        const bool have_next = (c + 1) < nch;

        // Prefetch next chunk's indices + scale/shift while chunk c is in flight.
        int    idx_nxt = 0;
        float2 ss_nxt  = {0.f, 0.f};
        int    m1      = 0;
        if (have_next) {
            m1 = min(CH, n - (c + 1) * CH);
            if (lane < m1) {
                idx_nxt = indices[start + (c + 1) * CH + lane];
                ss_nxt  = ssbase[idx_nxt];
            }
        }

        // Chunk c is fully landed in LDS once ASYNCcnt drains.
        asm volatile("s_wait_asynccnt 0" ::: "memory");

        // Kick off chunk c+1 into the other buffer before consuming chunk c,
        // so its 8KB of gathers overlap the dequant/accumulate below.
        if (have_next) {
            const unsigned bufoff = ((unsigned)(c + 1) & 1u) * (unsigned)(CH * ROWB);
            for (int j = 0; j < m1; ++j) {
                const int idxj = __shfl(idx_nxt, j, 32);
                const char* g  = qbase + (size_t)idxj * ROWB + lane * 16;
                const unsigned l = lds_wave + bufoff + (unsigned)(j * ROWB + lane * 16);
                asm volatile("global_load_async_to_lds_b128 %0, %1, off"
                             :: "v"(l), "v"(g) : "memory");
            }
        }

        // Consume chunk c: ds_load_b128 per row, rowwise dequant-accumulate.
        const int4* buf = smem + (size_t)(wv * 2 * CH + (c & 1) * CH) * (ROWB / 16);
#pragma unroll 4
        for (int j = 0; j < m; ++j) {
            const int4  q  = buf[j * 32 + lane];
            const float sc = __shfl(ss_cur.x, j, 32);
            const float sh = __shfl(ss_cur.y, j, 32);
            acc.x = fmaf((float)q.x, sc, acc.x);
            acc.y = fmaf((float)q.y, sc, acc.y);
            acc.z = fmaf((float)q.z, sc, acc.z);
            acc.w = fmaf((float)q.w, sc, acc.w);
            shift_sum += sh;
        }

        idx_cur = idx_nxt;
        ss_cur  = ss_nxt;
    }

    // shift is rank-1 across dims: add the per-bag shift sum to every element.
    acc.x += shift_sum;
    acc.y += shift_sum;
    acc.z += shift_sum;
    acc.w += shift_sum;

    // KeyedTensor layout: out[b, t*D + d]
    float4* o = (float4*)(out + (size_t)b * (TT * DD) + (size_t)t * DD);
    o[lane] = acc;
}

extern "C" void kernel_launch(void* const* d_in, const int* in_sizes, int n_in,
                              void* d_out, int out_size, void* d_ws, size_t ws_size,
                              hipStream_t stream) {
    const int*   indices     = (const int*)d_in[0];
    const int*   offsets     = (const int*)d_in[1];
    const int*   qweights    = (const int*)d_in[2];
    const float* scale_shift = (const float*)d_in[3];
    float*       out         = (float*)d_out;

    const int nbags = TT * BB;                 // 16384 bags, one wave each
    dim3 grid(nbags / WAVES), block(WAVES * 32);
    hipLaunchKernelGGL(tbe_int8_pool_kernel, grid, block, 0, stream,
                       indices, offsets, qweights, scale_shift, out);
    (void)in_sizes; (void)n_in; (void)out_size; (void)d_ws; (void)ws_size;
}